// DifferentiableFeatureLayer_87634512708042
// MI455X (gfx1250) — compile-verified
//
#include <hip/hip_runtime.h>
#include <math.h>

// Problem constants (from reference)
#define B_ 16
#define T_ 512
#define C_ 32
#define W_ 128
#define KSTEPS 36          // K = 144 = 36 x (K-slab of 4)
#define TAB_N 160          // weight table: index = w + 16, w in [-16, 144), zero guards

static constexpr float WIN_MIN_ = 2.0f;
static constexpr float WIN_MAX_ = 64.0f;

typedef __attribute__((ext_vector_type(2))) float v2f;
typedef __attribute__((ext_vector_type(8))) float v8f;

// sigmoid via v_exp_f32 + v_rcp_f32 (no IEEE division macro expansion)
__device__ __forceinline__ float sigmoidf_(float x) {
  return __builtin_amdgcn_rcpf(1.0f + __expf(-x));
}

// ---------------------------------------------------------------------------
// Kernel 1: WMMA sliding-window weighted sums.
// One wave (32 threads) per (channel, 16-wide t tile).
// GEMM view: M = batch (16), N = t-offset j (16), K = j + w in [0,144).
//   A[b,k]  = series[starts[b]-W+t0+k, c]                       (f32)
//   B[k,j]  = wtab[(k-j)+16]   (Toeplitz; zero-padded LDS table)
// D1 = A*B1, D2 = A*B2, D3 = (A^2)*B2 accumulated via v_wmma_f32_16x16x4_f32.
// ---------------------------------------------------------------------------
__global__ __launch_bounds__(32) void feat_wmma(
    const float* __restrict__ series,   // [SERIES_LEN, C]
    const int*   __restrict__ indices,  // [B, 2]
    const float* __restrict__ rwm,      // [C]
    const float* __restrict__ rws,      // [C]
    float* __restrict__ f_mean,         // [B,T,C] ws
    float* __restrict__ f_std)          // [B,T,C] ws
{
  __shared__ float w1tab[TAB_N];
  __shared__ float w2tab[TAB_N];

  const int c    = blockIdx.x;          // channel
  const int t0   = blockIdx.y << 4;     // t tile base
  const int lane = threadIdx.x;         // 0..31

  // ---- one-time per block: soft-window weight tables + normalizers ----
  const float win1 = WIN_MIN_ + sigmoidf_(rwm[c]) * (WIN_MAX_ - WIN_MIN_);
  const float win2 = WIN_MIN_ + sigmoidf_(rws[c]) * (WIN_MAX_ - WIN_MIN_);

  float p1 = 0.0f, p2 = 0.0f;
  #pragma unroll
  for (int i = lane; i < TAB_N; i += 32) {      // exactly 5 trips, no divergence
    const int w = i - 16;
    const float msk = ((w >= 0) & (w < W_)) ? 1.0f : 0.0f;  // branch-free select
    const float age = (float)(W_ - 1 - w);
    const float v1 = msk * sigmoidf_(win1 - age);
    const float v2 = msk * sigmoidf_(win2 - age);
    w1tab[i] = v1;
    w2tab[i] = v2;
    p1 += v1;
    p2 += v2;
  }
  // wave32 sum reduction -> s1, s2 (each in-range w counted exactly once)
  #pragma unroll
  for (int off = 16; off > 0; off >>= 1) {
    p1 += __shfl_xor(p1, off, 32);
    p2 += __shfl_xor(p2, off, 32);
  }
  __syncthreads();
  const float s1 = p1, s2 = p2;

  // ---- WMMA operand addressing ----
  const int m     = lane & 15;          // A-row (batch) this lane feeds
  const int khalf = (lane >> 4) << 1;   // K-pair selector per half-wave
  const int j     = m;                  // B-matrix column for this lane

  const int start_m = indices[2 * m];                 // starts[b=m]
  const int baseT   = start_m - W_ + t0;              // >= 0 since starts >= W
  const float* __restrict__ gA = series + (size_t)baseT * C_ + c;
  const int tb = khalf - j + 16;                      // table base (always >= 1)

  v8f d1 = {}; v8f d2 = {}; v8f d3 = {};

  #pragma unroll
  for (int s = 0; s < KSTEPS; ++s) {
    const int k0 = (s << 2) + khalf;    // this lane's K pair: k0, k0+1
    const float a0 = gA[(size_t)k0 * C_];
    const float a1 = gA[(size_t)(k0 + 1) * C_];
    const int ti = tb + (s << 2);
    v2f A  = { a0, a1 };
    v2f A2 = { a0 * a0, a1 * a1 };
    v2f B1 = { w1tab[ti], w1tab[ti + 1] };
    v2f B2 = { w2tab[ti], w2tab[ti + 1] };
    // (neg_a, A, neg_b, B, c_mod, C, reuse_a, reuse_b)
    d1 = __builtin_amdgcn_wmma_f32_16x16x4_f32(false, A,  false, B1, (short)0, d1, false, false);
    d2 = __builtin_amdgcn_wmma_f32_16x16x4_f32(false, A,  false, B2, (short)0, d2, false, false);
    d3 = __builtin_amdgcn_wmma_f32_16x16x4_f32(false, A2, false, B2, (short)0, d3, false, false);
  }

  // D layout (f32 16x16): VGPR r, lanes 0-15 -> M=r, lanes 16-31 -> M=r+8; N = lane&15
  const float inv_s1 = __builtin_amdgcn_rcpf(s1);
  const float inv_s2 = __builtin_amdgcn_rcpf(s2);
  const int bBase = (lane >> 4) << 3;
  const int t     = t0 + (lane & 15);
  #pragma unroll
  for (int r = 0; r < 8; ++r) {
    const int b     = bBase + r;
    const float fm  = d1[r] * inv_s1;
    const float mu  = d2[r] * inv_s2;
    const float var = d3[r] * inv_s2 - mu * mu;
    const float fs  = sqrtf(fmaxf(var, 0.0f) + 1e-8f);
    const int o = (b * T_ + t) * C_ + c;
    f_mean[o] = fm;
    f_std[o]  = fs;
  }
}

// ---------------------------------------------------------------------------
// Kernel 2: per-channel batchnorm statistics over (B,T). One block per channel.
// ---------------------------------------------------------------------------
__global__ __launch_bounds__(256) void bn_stats(
    const float* __restrict__ f_mean,
    const float* __restrict__ f_std,
    float* __restrict__ stats)          // [C][4] = {mean_m, rstd_m, mean_s, rstd_s}
{
  const int c   = blockIdx.x;
  const int tid = threadIdx.x;
  float sm = 0.f, sm2 = 0.f, ss = 0.f, ss2 = 0.f;
  for (int i = tid; i < B_ * T_; i += 256) {
    const float vm = f_mean[i * C_ + c];
    const float vs = f_std [i * C_ + c];
    sm += vm; sm2 += vm * vm; ss += vs; ss2 += vs * vs;
  }
  __shared__ float red[4][256];
  red[0][tid] = sm; red[1][tid] = sm2; red[2][tid] = ss; red[3][tid] = ss2;
  __syncthreads();
  for (int off = 128; off > 0; off >>= 1) {
    if (tid < off) {
      red[0][tid] += red[0][tid + off];
      red[1][tid] += red[1][tid + off];
      red[2][tid] += red[2][tid + off];
      red[3][tid] += red[3][tid + off];
    }
    __syncthreads();
  }
  if (tid == 0) {
    const float n  = (float)(B_ * T_);
    const float mm = red[0][0] / n;
    const float vm = red[1][0] / n - mm * mm;
    const float ms = red[2][0] / n;
    const float vs = red[3][0] / n - ms * ms;
    stats[c * 4 + 0] = mm;
    stats[c * 4 + 1] = rsqrtf(fmaxf(vm, 0.0f) + 1e-5f);
    stats[c * 4 + 2] = ms;
    stats[c * 4 + 3] = rsqrtf(fmaxf(vs, 0.0f) + 1e-5f);
  }
}

// ---------------------------------------------------------------------------
// Kernel 3: normalize + concat [x, bn(f_mean), bn(f_std)] -> (B,T,96)
// ---------------------------------------------------------------------------
__global__ __launch_bounds__(256) void write_out(
    const float* __restrict__ x,
    const float* __restrict__ f_mean,
    const float* __restrict__ f_std,
    const float* __restrict__ stats,
    const float* __restrict__ gm, const float* __restrict__ bm,
    const float* __restrict__ gs, const float* __restrict__ bs,
    float* __restrict__ out)
{
  const int idx = blockIdx.x * 256 + threadIdx.x;
  if (idx >= B_ * T_ * C_) return;
  const int c  = idx & (C_ - 1);
  const int bt = idx / C_;
  out[bt * (3 * C_) + c] = x[idx];
  out[bt * (3 * C_) + C_ + c] =
      (f_mean[idx] - stats[c * 4 + 0]) * stats[c * 4 + 1] * gm[c] + bm[c];
  out[bt * (3 * C_) + 2 * C_ + c] =
      (f_std[idx]  - stats[c * 4 + 2]) * stats[c * 4 + 3] * gs[c] + bs[c];
}

extern "C" void kernel_launch(void* const* d_in, const int* in_sizes, int n_in,
                              void* d_out, int out_size, void* d_ws, size_t ws_size,
                              hipStream_t stream) {
  const float* x      = (const float*)d_in[0];
  const float* series = (const float*)d_in[1];
  const int*   indices= (const int*)  d_in[2];
  const float* rwm    = (const float*)d_in[3];
  const float* rws    = (const float*)d_in[4];
  const float* gm     = (const float*)d_in[5];
  const float* bm     = (const float*)d_in[6];
  const float* gs     = (const float*)d_in[7];
  const float* bs     = (const float*)d_in[8];
  float* out = (float*)d_out;

  float* f_mean = (float*)d_ws;                  // B*T*C floats
  float* f_std  = f_mean + (size_t)B_ * T_ * C_;
  float* stats  = f_std  + (size_t)B_ * T_ * C_; // C*4 floats

  feat_wmma<<<dim3(C_, T_ / 16), 32, 0, stream>>>(series, indices, rwm, rws,
                                                  f_mean, f_std);
  bn_stats<<<C_, 256, 0, stream>>>(f_mean, f_std, stats);
  write_out<<<(B_ * T_ * C_ + 255) / 256, 256, 0, stream>>>(
      x, f_mean, f_std, stats, gm, bm, gs, bs, out);
}